// GLU_16535624089675
// MI455X (gfx1250) — compile-verified
//
#include <hip/hip_runtime.h>

#define E_   8
#define FFN_ 8192
#define H_   2048
#define TOK_ 32
#define KSL_ 16   // split-K slices for the second GEMM

typedef __attribute__((ext_vector_type(2))) float v2f;
typedef __attribute__((ext_vector_type(8))) float v8f;

__device__ __forceinline__ v8f wmma_f32(v2f a, v2f b, v8f c) {
  // V_WMMA_F32_16X16X4_F32: D = A(16x4) * B(4x16) + C(16x16), all f32
  return __builtin_amdgcn_wmma_f32_16x16x4_f32(
      /*neg_a=*/false, a, /*neg_b=*/false, b,
      /*c_mod=*/(short)0, c, /*reuse_a=*/false, /*reuse_b=*/false);
}

__device__ __forceinline__ float fast_silu(float s) {
  // silu(s) = s * sigmoid(s); v_exp_f32 + v_rcp_f32, no IEEE div ladder
  return s * __builtin_amdgcn_rcpf(1.0f + __expf(-s));
}

// Kernel 1: g = silu(x @ ew1^T) * (x @ ev1^T)   -> g[TOK, FFN] in workspace
// One wave per (mTile in 0..1, nTile in 0..511): 1024 wave-tasks.
__global__ void glu_gate_kernel(const float* __restrict__ x,
                                const float* __restrict__ w1,
                                const float* __restrict__ v1,
                                const int*   __restrict__ eidx,
                                float* __restrict__ g) {
  const size_t eoff = (size_t)(*eidx) * (size_t)FFN_ * (size_t)H_;
  const float* W1 = w1 + eoff;
  const float* V1 = v1 + eoff;

  const int wave  = (int)((blockIdx.x * blockDim.x + threadIdx.x) >> 5);
  const int lane  = (int)(threadIdx.x & 31);
  const int mTile = wave & 1;         // which 16 tokens
  const int nTile = wave >> 1;        // which 16 ffn rows (0..511)
  const int nm    = lane & 15;        // A: M row / B: N col
  const int half  = lane >> 4;        // K sub-offset selector

  const float* xrow  = x  + (size_t)(mTile * 16 + nm) * H_ + 2 * half;
  const float* w1row = W1 + (size_t)(nTile * 16 + nm) * H_ + 2 * half;
  const float* v1row = V1 + (size_t)(nTile * 16 + nm) * H_ + 2 * half;

  v8f acc1 = {};
  v8f acc2 = {};
#pragma unroll 8
  for (int k = 0; k < H_; k += 4) {
    v2f a  = *(const v2f*)(xrow  + k);
    v2f b1 = *(const v2f*)(w1row + k);
    v2f b2 = *(const v2f*)(v1row + k);
    acc1 = wmma_f32(a, b1, acc1);
    acc2 = wmma_f32(a, b2, acc2);
  }

  // D layout: VGPR r -> token row (r + 8*half) within the M tile, col = nm
  const int tokBase = mTile * 16 + 8 * half;
  const int f       = nTile * 16 + nm;
#pragma unroll
  for (int r = 0; r < 8; ++r) {
    float val = fast_silu(acc1[r]) * acc2[r];
    g[(size_t)(tokBase + r) * FFN_ + f] = val;
  }
}

// Kernel 2: partial[ks] = g[:, kslice] @ ew2[kslice, :]
// One wave per (kslice 0..15, nTile 0..127, mTile 0..1): 4096 wave-tasks.
__global__ void down_proj_partial_kernel(const float* __restrict__ g,
                                         const float* __restrict__ w2,
                                         const int*   __restrict__ eidx,
                                         float* __restrict__ part) {
  const size_t eoff = (size_t)(*eidx) * (size_t)FFN_ * (size_t)H_;
  const float* W2 = w2 + eoff;

  const int wave  = (int)((blockIdx.x * blockDim.x + threadIdx.x) >> 5);
  const int lane  = (int)(threadIdx.x & 31);
  const int mTile = wave & 1;
  const int nTile = (wave >> 1) & 127;   // H/16 tiles
  const int ks    = wave >> 8;           // 0..15 k-slices
  const int nm    = lane & 15;
  const int half  = lane >> 4;
  const int col   = nTile * 16 + nm;

  const float* grow  = g  + (size_t)(mTile * 16 + nm) * FFN_ + 2 * half;
  const float* wbase = W2 + (size_t)(2 * half) * H_ + col;

  v8f acc = {};
  const int k0 = ks * (FFN_ / KSL_);
#pragma unroll 8
  for (int k = k0; k < k0 + FFN_ / KSL_; k += 4) {
    v2f a = *(const v2f*)(grow + k);
    v2f b;
    b.x = wbase[(size_t)k * H_];           // B VGPR0: K = k + 2*half
    b.y = wbase[(size_t)(k + 1) * H_];     // B VGPR1: K = k + 1 + 2*half
    acc = wmma_f32(a, b, acc);
  }

  const int tokBase = mTile * 16 + 8 * half;
  float* p = part + (size_t)ks * TOK_ * H_;
#pragma unroll
  for (int r = 0; r < 8; ++r)
    p[(size_t)(tokBase + r) * H_ + col] = acc[r];
}

// Kernel 3: deterministic fixed-order split-K reduction into d_out.
__global__ void reduce_kernel(const float* __restrict__ part,
                              float* __restrict__ out) {
  const int i = (int)(blockIdx.x * blockDim.x + threadIdx.x);
  if (i < TOK_ * H_) {
    float s = 0.0f;
#pragma unroll
    for (int ks = 0; ks < KSL_; ++ks)
      s += part[(size_t)ks * TOK_ * H_ + i];
    out[i] = s;
  }
}

extern "C" void kernel_launch(void* const* d_in, const int* in_sizes, int n_in,
                              void* d_out, int out_size, void* d_ws, size_t ws_size,
                              hipStream_t stream) {
  (void)in_sizes; (void)n_in; (void)out_size; (void)ws_size;
  const float* x    = (const float*)d_in[0];
  const float* w1   = (const float*)d_in[1];
  const float* v1   = (const float*)d_in[2];
  const float* w2   = (const float*)d_in[3];
  const int*   eidx = (const int*)d_in[4];
  float* out = (float*)d_out;

  float* g    = (float*)d_ws;                    // TOK*FFN floats   = 1 MB
  float* part = g + (size_t)TOK_ * FFN_;         // KSL*TOK*H floats = 4 MB

  // 1024 wave-tasks / 8 waves per 256-thread block -> 128 blocks
  glu_gate_kernel<<<128, 256, 0, stream>>>(x, w1, v1, eidx, g);
  // 4096 wave-tasks -> 512 blocks
  down_proj_partial_kernel<<<512, 256, 0, stream>>>(g, w2, eidx, part);
  // TOK*H = 65536 elements
  reduce_kernel<<<(TOK_ * H_ + 255) / 256, 256, 0, stream>>>(part, out);
}